// DownSampleBlock_661424964109
// MI455X (gfx1250) — compile-verified
//
#include <hip/hip_runtime.h>
#include <float.h>

typedef __attribute__((ext_vector_type(2))) float v2f;
typedef __attribute__((ext_vector_type(4))) float v4f;
typedef __attribute__((ext_vector_type(8))) float v8f;

#define KVOL 27
#define COUT 64
#define BN_EPS 1e-5f

// ---------------------------------------------------------------------------
// helpers
// ---------------------------------------------------------------------------
__device__ __forceinline__ void atomicMaxFloat(float* addr, float v) {
  // Sign-split trick: IEEE754 bit patterns are order-isomorphic per sign.
  if (v >= 0.0f) {
    atomicMax((int*)addr, __float_as_int(v));
  } else {
    atomicMin((unsigned int*)addr, __float_as_uint(v));
  }
}

// ---------------------------------------------------------------------------
// init: zero BN stats, pooled output = -FLT_MAX (every call, for determinism)
// ---------------------------------------------------------------------------
__global__ void init_kernel(double* __restrict__ stats, float* __restrict__ pooled,
                            int pooled_elems) {
  int i = blockIdx.x * blockDim.x + threadIdx.x;
  if (i < 256) stats[i] = 0.0;
  if (i < pooled_elems) pooled[i] = -FLT_MAX;
}

// ---------------------------------------------------------------------------
// Sparse conv: gather -> WMMA GEMM -> write pre-BN y + block-reduced BN stats.
// One wave = 16 voxels x 64 couts. Block = 8 waves = 128 voxels.
// Per kernel offset k, W[k] (CIN x 64 f32) is staged in LDS by the block.
// ---------------------------------------------------------------------------
template <int CIN>
__global__ __launch_bounds__(256) void sparse_conv_wmma(
    const float* __restrict__ fin,   // [N, CIN] (pad handled via idx==N -> 0)
    const float* __restrict__ W,     // [27, CIN, 64]
    const int* __restrict__ nbr,     // [N, 27], pad index == N
    float* __restrict__ yout,        // [N, 64] pre-BN output
    double* __restrict__ stats,      // [128]: sum[64], sumsq[64]
    int N) {
  __shared__ float ldsW[CIN * COUT];
  __shared__ float ldsStats[2 * COUT];

  const int tid = threadIdx.x;
  const int lane = tid & 31;
  const int wave = tid >> 5;
  const int tile = blockIdx.x * 8 + wave;
  const int base = tile * 16;
  const bool haveTile = (base < N);  // wave-uniform: EXEC stays all-1s for WMMA

  if (tid < 2 * COUT) ldsStats[tid] = 0.0f;

  const int m = lane & 15;                 // row within 16-voxel tile / col within 16-cout tile
  const int koff = (lane < 16) ? 0 : 2;    // K sub-pair for 32-bit A/B fragments
  const int row = base + m;
  const bool rowOk = (row < N);

  v8f acc[4] = {v8f{}, v8f{}, v8f{}, v8f{}};

  for (int k = 0; k < KVOL; ++k) {
    __syncthreads();  // previous k's consumers done (also covers ldsStats init)
    // cooperative stage of W[k] into LDS (b128 loads)
    {
      const v4f* wsrc = (const v4f*)(W + (size_t)k * (CIN * COUT));
      v4f* wdst = (v4f*)ldsW;
#pragma unroll
      for (int i = tid; i < (CIN * COUT) / 4; i += 256) wdst[i] = wsrc[i];
    }
    __syncthreads();

    if (haveTile) {
      int idx = rowOk ? nbr[(size_t)row * KVOL + k] : N;
      const bool valid = (idx < N);
      const float* src = fin + (size_t)idx * CIN + koff;
#pragma unroll
      for (int c = 0; c < CIN / 4; ++c) {
        // A fragment: lane holds row m, K = 4c+koff .. 4c+koff+1 (8B load)
        v2f a = {};
        if (valid) a = *(const v2f*)(src + c * 4);
        const int kk = c * 4 + koff;
        const float* wl = ldsW + kk * COUT + m;
        // B fragments for the 4 cout tiles: lane holds col n=t*16+m,
        // VGPR0 -> K=kk, VGPR1 -> K=kk+1
        v2f b0, b1, b2, b3;
        b0.x = wl[0];  b0.y = wl[COUT + 0];
        b1.x = wl[16]; b1.y = wl[COUT + 16];
        b2.x = wl[32]; b2.y = wl[COUT + 32];
        b3.x = wl[48]; b3.y = wl[COUT + 48];
        acc[0] = __builtin_amdgcn_wmma_f32_16x16x4_f32(false, a, false, b0, (short)0, acc[0], false, false);
        acc[1] = __builtin_amdgcn_wmma_f32_16x16x4_f32(false, a, false, b1, (short)0, acc[1], false, false);
        acc[2] = __builtin_amdgcn_wmma_f32_16x16x4_f32(false, a, false, b2, (short)0, acc[2], false, false);
        acc[3] = __builtin_amdgcn_wmma_f32_16x16x4_f32(false, a, false, b3, (short)0, acc[3], false, false);
      }
    }
  }

  // Epilogue: D layout — VGPR j: lanes0-15 -> row base+j, lanes16-31 -> row base+8+j
  if (haveTile) {
    const int rbase = base + ((lane < 16) ? 0 : 8);
#pragma unroll
    for (int t = 0; t < 4; ++t) {
      const int n = t * 16 + m;
      float s = 0.0f, s2 = 0.0f;
#pragma unroll
      for (int j = 0; j < 8; ++j) {
        const int r = rbase + j;
        const float v = acc[t][j];
        if (r < N) {
          yout[(size_t)r * COUT + n] = v;
          s += v;
          s2 += v * v;
        }
      }
      atomicAdd(&ldsStats[n], s);          // ds_add_f32
      atomicAdd(&ldsStats[COUT + n], s2);  // ds_add_f32
    }
  }
  __syncthreads();
  if (tid < 2 * COUT) atomicAdd(&stats[tid], (double)ldsStats[tid]);  // f64 global atomic
}

// ---------------------------------------------------------------------------
// BN (+ReLU) elementwise pass: x = relu((y - mean) * rsqrt(var+eps) * g + b)
// ---------------------------------------------------------------------------
__global__ void bn_relu_kernel(const float* __restrict__ y, float* __restrict__ x,
                               const double* __restrict__ stats,
                               const float* __restrict__ gamma,
                               const float* __restrict__ beta, int N) {
  const int i = blockIdx.x * blockDim.x + threadIdx.x;
  if (i >= N * COUT) return;
  const int c = i & (COUT - 1);
  const double invN = 1.0 / (double)N;
  const double mean = stats[c] * invN;
  const double var = stats[COUT + c] * invN - mean * mean;
  const float scale = gamma[c] * rsqrtf((float)var + BN_EPS);
  const float shift = beta[c] - (float)mean * scale;
  x[i] = fmaxf(fmaf(y[i], scale, shift), 0.0f);
}

// ---------------------------------------------------------------------------
// BN (no ReLU) + fused segment-max pooling
// ---------------------------------------------------------------------------
__global__ void bn_pool_kernel(const float* __restrict__ y, float* __restrict__ x,
                               const double* __restrict__ stats,
                               const float* __restrict__ gamma,
                               const float* __restrict__ beta,
                               const int* __restrict__ pool_ids,
                               float* __restrict__ pooled, int N) {
  const int i = blockIdx.x * blockDim.x + threadIdx.x;
  if (i >= N * COUT) return;
  const int c = i & (COUT - 1);
  const int r = i >> 6;
  const double invN = 1.0 / (double)N;
  const double mean = stats[c] * invN;
  const double var = stats[COUT + c] * invN - mean * mean;
  const float scale = gamma[c] * rsqrtf((float)var + BN_EPS);
  const float shift = beta[c] - (float)mean * scale;
  const float v = fmaf(y[i], scale, shift);
  x[i] = v;
  const int pid = pool_ids[r];
  atomicMaxFloat(&pooled[(size_t)pid * COUT + c], v);
}

// ---------------------------------------------------------------------------
// launch
// ---------------------------------------------------------------------------
extern "C" void kernel_launch(void* const* d_in, const int* in_sizes, int n_in,
                              void* d_out, int out_size, void* d_ws, size_t ws_size,
                              hipStream_t stream) {
  const float* feats  = (const float*)d_in[0];
  const float* W1     = (const float*)d_in[1];
  const float* gamma1 = (const float*)d_in[2];
  const float* beta1  = (const float*)d_in[3];
  const float* W2     = (const float*)d_in[4];
  const float* gamma2 = (const float*)d_in[5];
  const float* beta2  = (const float*)d_in[6];
  const int*   nbr    = (const int*)d_in[7];
  const int*   pids   = (const int*)d_in[8];
  // d_in[9] = num_pooled (device scalar) — recovered from out_size instead.

  const int N = in_sizes[0] / 32;
  const int num_pooled = out_size / COUT - N;

  float* x_out  = (float*)d_out;                  // [N, 64]
  float* pooled = x_out + (size_t)N * COUT;       // [num_pooled, 64]

  float*  y1     = (float*)d_ws;                  // [N, 64] scratch (y1 -> x1 in place)
  double* stats  = (double*)((char*)d_ws + (size_t)N * COUT * sizeof(float));
  double* stats1 = stats;                         // [128]
  double* stats2 = stats + 128;                   // [128]

  int initElems = num_pooled * COUT;
  if (initElems < 256) initElems = 256;
  init_kernel<<<(initElems + 255) / 256, 256, 0, stream>>>(stats, pooled,
                                                           num_pooled * COUT);

  const int tiles = (N + 15) / 16;
  const int convBlocks = (tiles + 7) / 8;
  const int elems = N * COUT;
  const int ewBlocks = (elems + 255) / 256;

  // conv1 (Cin=32) -> y1 (+stats1), then BN+ReLU in place -> x1
  sparse_conv_wmma<32><<<convBlocks, 256, 0, stream>>>(feats, W1, nbr, y1, stats1, N);
  bn_relu_kernel<<<ewBlocks, 256, 0, stream>>>(y1, y1, stats1, gamma1, beta1, N);

  // conv2 (Cin=64) -> pre-BN y2 directly into d_out (+stats2)
  sparse_conv_wmma<64><<<convBlocks, 256, 0, stream>>>(y1, W2, nbr, x_out, stats2, N);

  // BN in place + fused segment-max pooling
  bn_pool_kernel<<<ewBlocks, 256, 0, stream>>>(x_out, x_out, stats2, gamma2, beta2,
                                               pids, pooled, N);
}